// TopKGate_2834678415770
// MI455X (gfx1250) — compile-verified
//
#include <hip/hip_runtime.h>
#include <hip/hip_bf16.h>
#include <math.h>

typedef __attribute__((ext_vector_type(2))) float v2f;
typedef __attribute__((ext_vector_type(4))) float v4f;
typedef __attribute__((ext_vector_type(8))) float v8f;

#define S_TOK 8192
#define D_DIM 1024
#define E_EXP 64
#define CAP   128
#define TPB_TOK 64                 // tokens per block in gemm/scatter kernels
#define NBLK (S_TOK / TPB_TOK)     // 128 blocks
#define LROW 68                    // padded LDS row stride (bank-conflict free)

// ---------------------------------------------------------------------------
// Kernel 1: WMMA fp32 GEMM (logits = X @ W^T) fused with softmax/argmax gate.
// Block = 128 threads = 4 waves; each wave computes a 16x64 logits tile with
// V_WMMA_F32_16X16X4_F32 (exact fp32 math, matching the fp32 reference gate).
// ---------------------------------------------------------------------------
__global__ __launch_bounds__(128)
void moe_gate_gemm_kernel(const float* __restrict__ X,
                          const float* __restrict__ W,
                          int*   __restrict__ idx_out,     // [S]
                          float* __restrict__ gval_out,    // [S] gates[s, argmax]
                          int*   __restrict__ blkCounts,   // [NBLK][E]
                          float* __restrict__ blkGateSum)  // [NBLK][E]
{
    __shared__ float lds_logits[TPB_TOK * LROW];
    __shared__ int   lds_cnt[E_EXP];
    __shared__ float lds_max[TPB_TOK];
    __shared__ float lds_inv[TPB_TOK];

    const int tid  = threadIdx.x;
    const int wave = tid >> 5;
    const int lane = tid & 31;
    const int l16  = lane & 15;
    const int hi   = lane >> 4;          // 0: K+{0,1}, 1: K+{2,3}

    const int tokBase = blockIdx.x * TPB_TOK + wave * 16;

    v8f acc0 = {}, acc1 = {}, acc2 = {}, acc3 = {};

    const float* xrow = X + (size_t)(tokBase + l16) * D_DIM;
    const float* w0   = W + (size_t)( 0 + l16) * D_DIM;
    const float* w1   = W + (size_t)(16 + l16) * D_DIM;
    const float* w2   = W + (size_t)(32 + l16) * D_DIM;
    const float* w3   = W + (size_t)(48 + l16) * D_DIM;

    #pragma unroll 4
    for (int k = 0; k < D_DIM; k += 4) {
        const int col = k + 2 * hi;
        // A 16x4: lane(M=l16) holds K = 2*hi + {0,1}  -> contiguous float2
        v2f a  = *(const v2f*)(xrow + col);
        // B 4x16: lane(N=l16) holds K = 2*hi + {0,1}; B[k][n] = W[n][k]
        v2f b0 = *(const v2f*)(w0 + col);
        v2f b1 = *(const v2f*)(w1 + col);
        v2f b2 = *(const v2f*)(w2 + col);
        v2f b3 = *(const v2f*)(w3 + col);
        acc0 = __builtin_amdgcn_wmma_f32_16x16x4_f32(false, a, false, b0, (short)0, acc0, false, false);
        acc1 = __builtin_amdgcn_wmma_f32_16x16x4_f32(false, a, false, b1, (short)0, acc1, false, false);
        acc2 = __builtin_amdgcn_wmma_f32_16x16x4_f32(false, a, false, b2, (short)0, acc2, false, false);
        acc3 = __builtin_amdgcn_wmma_f32_16x16x4_f32(false, a, false, b3, (short)0, acc3, false, false);
    }

    // C/D layout: VGPR r -> (m = r + 8*hi, n = l16). Token t row at t*LROW.
    float* myrows = lds_logits + wave * 16 * LROW;
    #pragma unroll
    for (int r = 0; r < 8; ++r) {
        const int m = r + 8 * hi;
        myrows[m * LROW +  0 + l16] = acc0[r];
        myrows[m * LROW + 16 + l16] = acc1[r];
        myrows[m * LROW + 32 + l16] = acc2[r];
        myrows[m * LROW + 48 + l16] = acc3[r];
    }
    if (tid < E_EXP) lds_cnt[tid] = 0;
    __syncthreads();

    // Phase 2: one thread per token -> max / first-argmax / softmax denom
    if (tid < TPB_TOK) {
        const float* row = lds_logits + tid * LROW;   // banks (4t+e)%64: conflict-free
        float best = row[0];
        int   bi   = 0;
        #pragma unroll 8
        for (int e = 1; e < E_EXP; ++e) {
            float v = row[e];
            if (v > best) { best = v; bi = e; }       // strict '>' == first-occurrence
        }
        float sum = 0.f;
        #pragma unroll 8
        for (int e = 0; e < E_EXP; ++e) sum += __expf(row[e] - best);
        const float inv = 1.0f / sum;                 // gates[s, bi] = exp(0)/sum
        const int s = blockIdx.x * TPB_TOK + tid;
        idx_out[s]  = bi;
        gval_out[s] = inv;
        lds_max[tid] = best;
        lds_inv[tid] = inv;
        atomicAdd(&lds_cnt[bi], 1);                   // order-independent int
    }
    __syncthreads();

    // Phase 3: one thread per expert -> deterministic per-block gate sums
    if (tid < E_EXP) {
        float gs = 0.f;
        for (int t = 0; t < TPB_TOK; ++t)
            gs += __expf(lds_logits[t * LROW + tid] - lds_max[t]) * lds_inv[t];
        blkGateSum[blockIdx.x * E_EXP + tid] = gs;
        blkCounts [blockIdx.x * E_EXP + tid] = lds_cnt[tid];
    }
}

// ---------------------------------------------------------------------------
// Kernel 2: zero-fill d_out with nontemporal b128 stores (512MB >> 192MB L2).
// ---------------------------------------------------------------------------
__global__ __launch_bounds__(256)
void moe_gate_fill_kernel(float* __restrict__ p, size_t n)
{
    const size_t i      = (size_t)blockIdx.x * blockDim.x + threadIdx.x;
    const size_t stride = (size_t)gridDim.x * blockDim.x;
    const size_t n4     = n >> 2;
    v4f z = {};
    for (size_t j = i; j < n4; j += stride)
        __builtin_nontemporal_store(z, (v4f*)p + j);
    if (i < (n & (size_t)3)) p[(n4 << 2) + i] = 0.f;
}

// ---------------------------------------------------------------------------
// Kernel 3: single-block sequential scan over block partials (deterministic):
// exclusive per-expert offsets, exp_counts, and l_aux.
// ---------------------------------------------------------------------------
__global__ __launch_bounds__(64)
void moe_gate_scan_kernel(const int*   __restrict__ blkCounts,
                          const float* __restrict__ blkGateSum,
                          int*   __restrict__ blkOffsets,
                          float* __restrict__ out)
{
    __shared__ float laux_part[E_EXP];
    const int e = threadIdx.x;
    int   running = 0;
    float gs      = 0.f;
    for (int b = 0; b < NBLK; ++b) {                  // fixed order => deterministic
        blkOffsets[b * E_EXP + e] = running;
        running += blkCounts [b * E_EXP + e];
        gs      += blkGateSum[b * E_EXP + e];
    }
    const size_t N = (size_t)S_TOK * E_EXP * CAP;
    out[1 + 2 * N + e] = (float)running;              // exp_counts (pre-capacity)
    const float me = gs / (float)S_TOK;
    const float ce = (float)running / (float)S_TOK;
    laux_part[e] = me * ce;
    __syncthreads();
    if (e == 0) {
        float acc = 0.f;
        for (int i = 0; i < E_EXP; ++i) acc += laux_part[i];
        out[0] = acc * (float)E_EXP;                  // l_aux
    }
}

// ---------------------------------------------------------------------------
// Kernel 4: sparse scatter of the 8192 nonzeros into combine/dispatch.
// Within-block rank (token order) + scanned block offset = position-in-expert.
// ---------------------------------------------------------------------------
__global__ __launch_bounds__(64)
void moe_gate_scatter_kernel(const int*   __restrict__ idx_in,
                             const float* __restrict__ gval,
                             const int*   __restrict__ blkOffsets,
                             float* __restrict__ out)
{
    __shared__ int lidx[TPB_TOK];
    const int t = threadIdx.x;
    const int s = blockIdx.x * TPB_TOK + t;
    const int my = idx_in[s];
    lidx[t] = my;
    __syncthreads();
    int rank = 0;
    for (int u = 0; u < t; ++u) rank += (lidx[u] == my) ? 1 : 0;
    const int pos = blkOffsets[blockIdx.x * E_EXP + my] + rank;
    if (pos < CAP) {                                  // capacity keep/drop
        const size_t N    = (size_t)S_TOK * E_EXP * CAP;
        const size_t base = 1 + ((size_t)s * E_EXP + my) * CAP + pos;
        out[base]     = gval[s];                      // combine_weights
        out[base + N] = 1.0f;                         // dispatch_mask (as float)
    }
}

extern "C" void kernel_launch(void* const* d_in, const int* in_sizes, int n_in,
                              void* d_out, int out_size, void* d_ws, size_t ws_size,
                              hipStream_t stream) {
    const float* X = (const float*)d_in[0];   // [8192, 1024]
    const float* W = (const float*)d_in[1];   // [64, 1024]
    float* out = (float*)d_out;               // [1 + 2*S*E*CAP + E]

    // Workspace partition (160 KB total)
    int*   idx        = (int*)d_ws;                       // S
    float* gval       = (float*)(idx + S_TOK);            // S
    int*   blkCounts  = (int*)(gval + S_TOK);             // NBLK*E
    float* blkGateSum = (float*)(blkCounts + NBLK * E_EXP);
    int*   blkOffsets = (int*)(blkGateSum + NBLK * E_EXP);

    const size_t total = (size_t)1 + 2 * (size_t)S_TOK * E_EXP * CAP + E_EXP;

    // Long pole first: ~536MB nontemporal zero-fill at HBM rate.
    moe_gate_fill_kernel<<<4096, 256, 0, stream>>>(out, total);

    // WMMA GEMM + fused gate.
    moe_gate_gemm_kernel<<<NBLK, 128, 0, stream>>>(X, W, idx, gval,
                                                   blkCounts, blkGateSum);

    // Deterministic block scan -> offsets, exp_counts, l_aux.
    moe_gate_scan_kernel<<<1, 64, 0, stream>>>(blkCounts, blkGateSum,
                                               blkOffsets, out);

    // Sparse scatter of kept tokens.
    moe_gate_scatter_kernel<<<NBLK, 64, 0, stream>>>(idx, gval, blkOffsets, out);
}